// MeanAggregator_16226386444816
// MI455X (gfx1250) — compile-verified
//
#include <hip/hip_runtime.h>

// GraphSAGE mean-aggregator + dense layer, fused, for MI455X (gfx1250).
//
//   out[b, :] = relu(concat(features[node[b]], mean_j features[nbr[b,j]]) @ W)
//
// One workgroup = 512 threads (16 wave32) handles a 64-row tile of the batch.
// Phase A: gather node row + 25-neighbour mean into LDS X-tile (f32, padded).
//          W (256x128 f32) is staged once into LDS *K-major* so each WMMA
//          B-fragment is one contiguous ds_load_b64 (no register shuffles).
// Phase B: 16x16 output tiles via V_WMMA_F32_16X16X4_F32, K=256 in steps of 4.
//          f32 WMMA keeps full reference precision; the kernel is L2/gather
//          bound (~436 MB gather vs 2.15 GFLOP), so matrix-op rate is free.

typedef __attribute__((ext_vector_type(2))) float v2f;
typedef __attribute__((ext_vector_type(8))) float v8f;

#define DIM     128      // feature dim
#define KTOT    256      // concat dim (GEMM K)
#define UNITS   128      // output dim (GEMM N)
#define KNBR    25
#define M_TILE  64
#define XPITCH  260      // 256 + 4 pad -> A-frag: bank = (4*row + k) % 64, conflict-free
#define KP      258      // K-major W pitch: 256 + 2 pad, keeps 8B align, splits half-wave banks
#define NTHREADS 512

__global__ __launch_bounds__(NTHREADS, 1)
void sage_fused_wmma(const float* __restrict__ features,
                     const int*   __restrict__ node,
                     const int*   __restrict__ neigh,
                     const float* __restrict__ W,
                     float*       __restrict__ out,
                     int nrows)
{
    __shared__ float sX [M_TILE * XPITCH];  //  66,560 B
    __shared__ float sWt[UNITS  * KP];      // 132,096 B  (total < 320 KB WGP LDS)

    const int tid  = threadIdx.x;
    const int lane = tid & 31;
    const int wave = tid >> 5;
    const int rowBase = blockIdx.x * M_TILE;

    // ---- stage W into LDS, transposed to K-major: sWt[n*KP + k] = W[k][n] ----
    // global reads stay fully coalesced (float4 over n); transpose writes are
    // a one-time cost per block.
    for (int i4 = tid * 4; i4 < KTOT * UNITS; i4 += NTHREADS * 4) {
        const int k = i4 >> 7;          // /128
        const int c = i4 & 127;
        const float4 w4 = *(const float4*)(W + i4);
        sWt[(c + 0) * KP + k] = w4.x;
        sWt[(c + 1) * KP + k] = w4.y;
        sWt[(c + 2) * KP + k] = w4.z;
        sWt[(c + 3) * KP + k] = w4.w;
    }

    // ---- phase A: gather + neighbour mean into X-tile ----
    // one wave per row: 32 lanes x float4 = 512 B = one full feature row per vmem op
    const int c4 = lane * 4;
    for (int rr = 0; rr < M_TILE / 16; ++rr) {          // 4 rows per wave
        const int r = wave * 4 + rr;
        int g = rowBase + r;
        if (g >= nrows) g = nrows - 1;                  // clamp (B % 64 == 0 in practice)

        const int nid = node[g];
        const float4 self = *(const float4*)(features + (size_t)nid * DIM + c4);
        *(float4*)&sX[r * XPITCH + c4] = self;

        float4 acc = make_float4(0.f, 0.f, 0.f, 0.f);
        const int* nb = neigh + (size_t)g * KNBR;
        #pragma unroll 5
        for (int j = 0; j < KNBR; ++j) {
            const float4 v = *(const float4*)(features + (size_t)nb[j] * DIM + c4);
            acc.x += v.x; acc.y += v.y; acc.z += v.z; acc.w += v.w;
        }
        const float inv = 1.0f / (float)KNBR;
        acc.x *= inv; acc.y *= inv; acc.z *= inv; acc.w *= inv;
        *(float4*)&sX[r * XPITCH + DIM + c4] = acc;
    }

    __syncthreads();

    // ---- phase B: 4 M-tiles x 8 N-tiles over 16 waves (2 N-tiles each) ----
    const int m  = wave & 3;        // M-tile index within block
    const int ng = wave >> 2;       // N-tile-pair index (0..3)
    const int lm = lane & 15;
    const int kb = (lane >> 4) * 2; // f32 A/B frag: half-wave k-offset

    // A 16x4 frag: lane l -> A[l%16][kb..kb+1]  (one ds_load_b64 from sX)
    // B 4x16 frag: lane l -> B[kb..kb+1][l%16]  (one ds_load_b64 from K-major sWt)
    // C/D: vgpr v -> row v + (l/16)*8, col l%16.
    const float* ax = &sX[(m * 16 + lm) * XPITCH + kb];
    const float* bx0 = &sWt[((ng * 2 + 0) * 16 + lm) * KP + kb];
    const float* bx1 = &sWt[((ng * 2 + 1) * 16 + lm) * KP + kb];

    v8f acc0 = {};
    v8f acc1 = {};

    #pragma unroll 4
    for (int k = 0; k < KTOT; k += 4) {
        const v2f a  = *(const v2f*)&ax [k];
        const v2f b0 = *(const v2f*)&bx0[k];
        const v2f b1 = *(const v2f*)&bx1[k];
        acc0 = __builtin_amdgcn_wmma_f32_16x16x4_f32(
                   false, a, false, b0, (short)0, acc0, false, false);
        acc1 = __builtin_amdgcn_wmma_f32_16x16x4_f32(
                   false, a, false, b1, (short)0, acc1, false, false);
    }

    // ---- relu + store (half-wave writes 64 B contiguous per row) ----
    const int rhalf = (lane >> 4) * 8;
    const int gr0 = rowBase + m * 16 + rhalf;
    const size_t co0 = (size_t)(ng * 2 + 0) * 16 + lm;
    const size_t co1 = (size_t)(ng * 2 + 1) * 16 + lm;

    if (rowBase + M_TILE <= nrows) {
        // full tile: unguarded stores (common path; B % 64 == 0)
        #pragma unroll
        for (int v = 0; v < 8; ++v) {
            float x0 = acc0[v]; x0 = x0 > 0.f ? x0 : 0.f;
            float x1 = acc1[v]; x1 = x1 > 0.f ? x1 : 0.f;
            out[(size_t)(gr0 + v) * UNITS + co0] = x0;
            out[(size_t)(gr0 + v) * UNITS + co1] = x1;
        }
    } else {
        for (int v = 0; v < 8; ++v) {
            if (gr0 + v < nrows) {
                float x0 = acc0[v]; x0 = x0 > 0.f ? x0 : 0.f;
                float x1 = acc1[v]; x1 = x1 > 0.f ? x1 : 0.f;
                out[(size_t)(gr0 + v) * UNITS + co0] = x0;
                out[(size_t)(gr0 + v) * UNITS + co1] = x1;
            }
        }
    }
}

extern "C" void kernel_launch(void* const* d_in, const int* in_sizes, int n_in,
                              void* d_out, int out_size, void* d_ws, size_t ws_size,
                              hipStream_t stream)
{
    const float* features = (const float*)d_in[0];   // (100000, 128) f32
    const int*   node     = (const int*)  d_in[1];   // (B, 1) i32
    const int*   neigh    = (const int*)  d_in[2];   // (B, 25) i32
    const float* W        = (const float*)d_in[3];   // (256, 128) f32
    float*       out      = (float*)d_out;           // (B, 128) f32

    const int B = in_sizes[1];                       // node has B elements
    const int nblocks = (B + M_TILE - 1) / M_TILE;

    sage_fused_wmma<<<nblocks, NTHREADS, 0, stream>>>(features, node, neigh, W, out, B);
}